// Model_48172353192667
// MI455X (gfx1250) — compile-verified
//
#include <hip/hip_runtime.h>
#include <math.h>

#define BB 64
#define JJ 25
#define VV 50000
#define NTILES (VV / 16)      // 3125, exact
#define KPAIRS 14             // K padded 25 -> 28, stored as float2 pairs

typedef float v2f __attribute__((ext_vector_type(2)));
typedef float v8f __attribute__((ext_vector_type(8)));

// ws layout (float2): ws2[b][kp][n], b<64, kp<14, n<16.
// ws2[b][kp][n] = ( Tt[b][2kp][n], Tt[b][2kp+1][n] ), Tt[b][j][r*4+c] = T[b,j,r,c] (rows 0..2), rest 0.
__device__ __forceinline__ int ws_idx(int b, int k, int n) {
    return (((b * KPAIRS + (k >> 1)) * 16 + n) << 1) + (k & 1);
}

// ---------------- Phase 1: pose chain (1 block, thread b = batch b) ----------------
__global__ void pose_kernel(const float* __restrict__ je,
                            const float* __restrict__ lp,
                            const float* __restrict__ gpi,
                            const float* __restrict__ sc,
                            float* __restrict__ out,   // joint_out at offset 0
                            float* __restrict__ ws) {
    const int b = threadIdx.x;
    if (b >= BB) return;

    float G[3][4];
    G[0][0] = 1.f; G[0][1] = 0.f; G[0][2] = 0.f; G[0][3] = 0.f;
    G[1][0] = 0.f; G[1][1] = 1.f; G[1][2] = 0.f; G[1][3] = 0.f;
    G[2][0] = 0.f; G[2][1] = 0.f; G[2][2] = 1.f; G[2][3] = 0.f;

    for (int j = 0; j < JJ; ++j) {
        const float ex = je[(b * JJ + j) * 3 + 0];
        const float ey = je[(b * JJ + j) * 3 + 1];
        const float ez = je[(b * JJ + j) * 3 + 2];
        const float cx = cosf(ex), sx = sinf(ex);
        const float cy = cosf(ey), sy = sinf(ey);
        const float cz = cosf(ez), sz = sinf(ez);
        float R[3][3];
        R[0][0] = cz * cy; R[0][1] = cz * sy * sx - sz * cx; R[0][2] = cz * sy * cx + sz * sx;
        R[1][0] = sz * cy; R[1][1] = sz * sy * sx + cz * cx; R[1][2] = sz * sy * cx - cz * sx;
        R[2][0] = -sy;     R[2][1] = cy * sx;                R[2][2] = cy * cx;

        // Lm = (local_pose with trans += skel_corr) @ rot4   (rows 0..2; row 3 = 0001)
        float Lm[3][4];
        #pragma unroll
        for (int r = 0; r < 3; ++r) {
            const float l0 = lp[j * 16 + r * 4 + 0];
            const float l1 = lp[j * 16 + r * 4 + 1];
            const float l2 = lp[j * 16 + r * 4 + 2];
            const float l3 = lp[j * 16 + r * 4 + 3] + sc[j * 3 + r];
            #pragma unroll
            for (int c = 0; c < 3; ++c)
                Lm[r][c] = l0 * R[0][c] + l1 * R[1][c] + l2 * R[2][c];
            Lm[r][3] = l3;
        }

        if (j > 0) {   // root (j==0): G stays identity per reference
            float Gn[3][4];
            #pragma unroll
            for (int r = 0; r < 3; ++r)
                #pragma unroll
                for (int c = 0; c < 4; ++c)
                    Gn[r][c] = G[r][0] * Lm[0][c] + G[r][1] * Lm[1][c] + G[r][2] * Lm[2][c]
                             + (c == 3 ? G[r][3] : 0.f);
            #pragma unroll
            for (int r = 0; r < 3; ++r)
                #pragma unroll
                for (int c = 0; c < 4; ++c) G[r][c] = Gn[r][c];
        }

        // joint_out
        out[(b * JJ + j) * 3 + 0] = G[0][3];
        out[(b * JJ + j) * 3 + 1] = G[1][3];
        out[(b * JJ + j) * 3 + 2] = G[2][3];

        // T = G @ global_pose_inv[j]  (gpi row 3 = 0001)
        float T[3][4];
        #pragma unroll
        for (int r = 0; r < 3; ++r)
            #pragma unroll
            for (int c = 0; c < 4; ++c)
                T[r][c] = G[r][0] * gpi[j * 16 + 0 * 4 + c]
                        + G[r][1] * gpi[j * 16 + 1 * 4 + c]
                        + G[r][2] * gpi[j * 16 + 2 * 4 + c]
                        + (c == 3 ? G[r][3] : 0.f);

        #pragma unroll
        for (int n = 0; n < 16; ++n)
            ws[ws_idx(b, j, n)] = (n < 12) ? T[n >> 2][n & 3] : 0.f;
    }
    // zero K padding rows 25..27
    for (int k = JJ; k < 2 * KPAIRS; ++k)
        for (int n = 0; n < 16; ++n) ws[ws_idx(b, k, n)] = 0.f;
}

// ---------------- Phase 2: WMMA skinning ----------------
__global__ void __launch_bounds__(128) skin_kernel(const float* __restrict__ vpos,
                                                   const float* __restrict__ w,
                                                   const float* __restrict__ pc,
                                                   const float* __restrict__ ic,
                                                   const float* __restrict__ wsT,
                                                   float* __restrict__ out) {
    __shared__ float sP[4][16][17];   // per-wave D tile staging, padded vs bank conflicts

    const int lane = threadIdx.x & 31;
    const int wv   = threadIdx.x >> 5;
    const int tile = blockIdx.x * 4 + wv;
    if (tile >= NTILES) return;       // whole-wave exit; no block-level barriers used

    const int m  = lane & 15;         // row (A) / col (B,D) index
    const int hi = lane >> 4;
    const int vrow = tile * 16 + m;

    // A operand: 16x4 f32 chunks of W, identical for all 64 batches.
    // Lane layout: VGPR pair = K { k0+2*hi, k0+2*hi+1 } for row m.
    v2f a[7];
    #pragma unroll
    for (int kc = 0; kc < 7; ++kc) {
        const int j0 = kc * 4 + 2 * hi;
        const float ax = (j0     < JJ) ? w[vrow * JJ + j0]     : 0.f;
        const float ay = (j0 + 1 < JJ) ? w[vrow * JJ + j0 + 1] : 0.f;
        a[kc].x = ax; a[kc].y = ay;
    }

    // Base vertex position (batch-independent), used by epilogue lanes 0..15.
    float bx = 0.f, by = 0.f, bz = 0.f;
    if (hi == 0) {
        bx = vpos[vrow * 3 + 0];
        by = vpos[vrow * 3 + 1];
        bz = vpos[vrow * 3 + 2];
    }

    const float2* __restrict__ wsp = (const float2*)wsT;

    for (int bt = 0; bt < BB; ++bt) {
        // B operand: Tt chunks, one b64 load per chunk (pair (Tt[k][m], Tt[k+1][m])).
        v8f c = {};
        #pragma unroll
        for (int kc = 0; kc < 7; ++kc) {
            const float2 t = wsp[(bt * KPAIRS + kc * 2 + hi) * 16 + m];
            v2f bb; bb.x = t.x; bb.y = t.y;
            c = __builtin_amdgcn_wmma_f32_16x16x4_f32(
                    false, a[kc], false, bb, (short)0, c, false, false);
        }

        // D tile -> LDS rows: VGPR i holds (row 8*hi+i, col m)
        #pragma unroll
        for (int i = 0; i < 8; ++i) sP[wv][hi * 8 + i][m] = c[i];

        asm volatile("s_wait_dscnt 0" ::: "memory");
        __builtin_amdgcn_wave_barrier();

        if (hi == 0) {  // lane m finishes vertex vrow for batch bt
            const float* __restrict__ p = &sP[wv][m][0];
            const int gi = (bt * VV + vrow) * 3;
            const float mx = bx + pc[gi + 0] + ic[gi + 0];
            const float my = by + pc[gi + 1] + ic[gi + 1];
            const float mz = bz + pc[gi + 2] + ic[gi + 2];
            const float o0 = p[0] * mx + p[1] * my + p[2]  * mz + p[3];
            const float o1 = p[4] * mx + p[5] * my + p[6]  * mz + p[7];
            const float o2 = p[8] * mx + p[9] * my + p[10] * mz + p[11];
            const size_t ob = (size_t)(BB * JJ * 3) + (size_t)gi;
            out[ob + 0] = o0; out[ob + 1] = o1; out[ob + 2] = o2;
        }
        __builtin_amdgcn_wave_barrier();
    }
}

extern "C" void kernel_launch(void* const* d_in, const int* in_sizes, int n_in,
                              void* d_out, int out_size, void* d_ws, size_t ws_size,
                              hipStream_t stream) {
    const float* je  = (const float*)d_in[0];   // joint_euler (B,J,3)
    const float* vp  = (const float*)d_in[1];   // v (V,3)
    const float* sw  = (const float*)d_in[2];   // skinning_weight (V,J)
    const float* lp  = (const float*)d_in[3];   // local_pose (J,4,4)
    const float* gpi = (const float*)d_in[4];   // global_pose_inv (J,4,4)
    const float* sc  = (const float*)d_in[5];   // skeleton_corrective (J,3)
    const float* pc  = (const float*)d_in[6];   // pose_corrective (B,V,3)
    const float* ic  = (const float*)d_in[7];   // id_corrective (B,V,3)
    (void)d_in[8];                              // parents: chain j-1 by construction
    (void)in_sizes; (void)n_in; (void)out_size; (void)ws_size;

    float* out = (float*)d_out;
    float* ws  = (float*)d_ws;   // needs 64*14*16*2*4 = 114688 bytes

    pose_kernel<<<1, BB, 0, stream>>>(je, lp, gpi, sc, out, ws);
    skin_kernel<<<(NTILES + 3) / 4, 128, 0, stream>>>(vp, sw, pc, ic, ws, out);
}